// SelfAttention_28226525069862
// MI455X (gfx1250) — compile-verified
//
#include <hip/hip_runtime.h>
#include <hip/hip_bf16.h>

typedef __attribute__((ext_vector_type(2))) float v2f;
typedef __attribute__((ext_vector_type(8))) float v8f;

// FakeQuantize: scale=0.1, zp=0, qmin=-128, qmax=127, round half-to-even (rintf = RNE)
__device__ __forceinline__ float fq(float x) {
    float q = rintf(x / 0.1f);
    q = fminf(fmaxf(q, -128.0f), 127.0f);
    return q * 0.1f;
}

// ---------------------------------------------------------------------------
// Kernel 1: fold QA-LoRA into effective weights.
//   W_eff[proj][o][c] = fq(w[o][c]) + (ALPHA/L) * (fq(dB)@fq(dW))[o][c/64]
// ---------------------------------------------------------------------------
__global__ __launch_bounds__(256)
void qalora_prep_weights(const float* __restrict__ w0, const float* __restrict__ dW0, const float* __restrict__ dB0,
                         const float* __restrict__ w1, const float* __restrict__ dW1, const float* __restrict__ dB1,
                         const float* __restrict__ w2, const float* __restrict__ dW2, const float* __restrict__ dB2,
                         const float* __restrict__ w3, const float* __restrict__ dW3, const float* __restrict__ dB3,
                         float* __restrict__ Weff)
{
    __shared__ float delta[16];
    const int proj = blockIdx.x >> 10;
    const int o    = blockIdx.x & 1023;
    const float* w  = (proj == 0) ? w0  : (proj == 1) ? w1  : (proj == 2) ? w2  : w3;
    const float* dW = (proj == 0) ? dW0 : (proj == 1) ? dW1 : (proj == 2) ? dW2 : dW3;
    const float* dB = (proj == 0) ? dB0 : (proj == 1) ? dB1 : (proj == 2) ? dB2 : dB3;
    const int t = threadIdx.x;

    if (t < 16) {  // delta row: (1 x 32) @ (32 x 16)
        float acc = 0.0f;
        for (int r = 0; r < 32; ++r)
            acc += fq(dB[o * 32 + r]) * fq(dW[r * 16 + t]);
        delta[t] = acc;
    }
    __syncthreads();

    float* out = Weff + (size_t)proj * 1048576u + (size_t)o * 1024u;
    const float* wr = w + (size_t)o * 1024u;
    for (int c = t; c < 1024; c += 256)
        out[c] = fq(wr[c]) + 0.03125f * delta[c >> 6];  // 2.0/64
}

// ---------------------------------------------------------------------------
// Kernel 2: Y = X @ W_eff^T via V_WMMA_F32_16X16X4_F32.
//   Block tile 256(M) x 64(N), BK=16, ping-pong LDS double buffer.
//   Each wave owns a 32x64 strip (2 M-subtiles x 4 N-tiles = 8 accumulators):
//   B-fragments are reused across both M-subtiles -> 32 WMMA per 24 ds loads.
// ---------------------------------------------------------------------------
__global__ __launch_bounds__(256)
void gemm_weff_wmma(const float* __restrict__ X, const float* __restrict__ Wbase,
                    float* __restrict__ Ybase, int out_mode)
{
    __shared__ float As[2][256][17];   // 17-stride: conflict-free frag reads
    __shared__ float Bs[2][64][17];

    const float* W = Wbase + (size_t)blockIdx.z * 1048576u;
    float*       Y = Ybase + (size_t)blockIdx.z * 4194304u;

    const int tid  = threadIdx.x;
    const int lane = tid & 31;
    const int wv   = tid >> 5;
    const int l16  = lane & 15;
    const int koff = (lane < 16) ? 0 : 2;
    const int hi   = (lane < 16) ? 0 : 8;
    const int mBase = blockIdx.y * 256;
    const int nBase = blockIdx.x * 64;

    // staging: thread owns 4 float4 of A (rows srow+64i) and 1 float4 of B
    const int srow = tid >> 2;          // 0..63
    const int scol = (tid & 3) * 4;     // 0,4,8,12
    const float* pA = &X[(size_t)(mBase + srow) * 1024u + scol];
    const float* pB = &W[(size_t)(nBase + srow) * 1024u + scol];

    float4 ra[4], rb;
    #pragma unroll
    for (int i = 0; i < 4; ++i)
        ra[i] = *(const float4*)(pA + (size_t)i * 64u * 1024u);
    rb = *(const float4*)pB;
    #pragma unroll
    for (int i = 0; i < 4; ++i) {
        int r = srow + i * 64;
        As[0][r][scol+0]=ra[i].x; As[0][r][scol+1]=ra[i].y;
        As[0][r][scol+2]=ra[i].z; As[0][r][scol+3]=ra[i].w;
    }
    Bs[0][srow][scol+0]=rb.x; Bs[0][srow][scol+1]=rb.y;
    Bs[0][srow][scol+2]=rb.z; Bs[0][srow][scol+3]=rb.w;
    __syncthreads();

    v8f acc[2][4] = {};
    int buf = 0;
    for (int k0 = 0; k0 < 1024; k0 += 16) {
        const bool notlast = (k0 + 16 < 1024);
        if (notlast) {   // prefetch next tile while computing this one
            #pragma unroll
            for (int i = 0; i < 4; ++i)
                ra[i] = *(const float4*)(pA + (size_t)i * 64u * 1024u + k0 + 16);
            rb = *(const float4*)(pB + k0 + 16);
        }

        // hoist all fragments for this stage, then issue WMMAs back-to-back
        v2f afr[2][4], bfr[4][4];
        #pragma unroll
        for (int kk = 0; kk < 4; ++kk) {
            #pragma unroll
            for (int ms = 0; ms < 2; ++ms) {
                afr[ms][kk].x = As[buf][wv * 32 + ms * 16 + l16][kk * 4 + koff];
                afr[ms][kk].y = As[buf][wv * 32 + ms * 16 + l16][kk * 4 + koff + 1];
            }
            #pragma unroll
            for (int t = 0; t < 4; ++t) {
                bfr[kk][t].x = Bs[buf][t * 16 + l16][kk * 4 + koff];
                bfr[kk][t].y = Bs[buf][t * 16 + l16][kk * 4 + koff + 1];
            }
        }
        #pragma unroll
        for (int kk = 0; kk < 4; ++kk)
            #pragma unroll
            for (int t = 0; t < 4; ++t)
                #pragma unroll
                for (int ms = 0; ms < 2; ++ms)
                    acc[ms][t] = __builtin_amdgcn_wmma_f32_16x16x4_f32(
                        false, afr[ms][kk], false, bfr[kk][t], (short)0, acc[ms][t], false, false);

        if (notlast) {
            const int nb = buf ^ 1;
            #pragma unroll
            for (int i = 0; i < 4; ++i) {
                int r = srow + i * 64;
                As[nb][r][scol+0]=ra[i].x; As[nb][r][scol+1]=ra[i].y;
                As[nb][r][scol+2]=ra[i].z; As[nb][r][scol+3]=ra[i].w;
            }
            Bs[nb][srow][scol+0]=rb.x; Bs[nb][srow][scol+1]=rb.y;
            Bs[nb][srow][scol+2]=rb.z; Bs[nb][srow][scol+3]=rb.w;
            __syncthreads();
            buf = nb;
        }
    }

    // C/D layout: VGPR v -> row v (lanes 0-15) / v+8 (lanes 16-31), col = l16
    #pragma unroll
    for (int ms = 0; ms < 2; ++ms) {
        #pragma unroll
        for (int t = 0; t < 4; ++t) {
            #pragma unroll
            for (int v = 0; v < 8; ++v) {
                int m = mBase + wv * 32 + ms * 16 + v + hi;
                int n = nBase + t * 16 + l16;
                float val = acc[ms][t][v];
                if (out_mode == 0) {
                    Y[(size_t)m * 1024u + n] = val;
                } else {
                    int b = m >> 11, nn = m & 2047;
                    int h = n >> 6,  d  = n & 63;
                    Y[(((size_t)(b * 16 + h) * 2048u + nn) << 6) + d] = val;
                }
            }
        }
    }
}

// ---------------------------------------------------------------------------
// Kernel 3: flash attention, fp32 WMMA for QK^T and PV.
//   Q/K/V in (bh, n, d), n=2048, d=64. 8 waves x 16 query rows = 128/block.
//   Ping-pong K/V staging; frag loads grouped per 4-kstep chunk.
//   V stride 80 so the two lane-halves (rows r and r+2) hit disjoint banks.
// ---------------------------------------------------------------------------
__global__ __launch_bounds__(256)
void flash_attn_wmma(const float* __restrict__ Q, const float* __restrict__ K,
                     const float* __restrict__ V, float* __restrict__ O)
{
    __shared__ float Ks[2][64][68];
    __shared__ float Vs[2][64][80];
    __shared__ float Ps[8][16][68];   // per-wave P relayout buffer

    const int bh = blockIdx.y;
    const int b  = bh >> 4, h = bh & 15;
    const int tid  = threadIdx.x;
    const int lane = tid & 31;
    const int wv   = tid >> 5;
    const int l16  = lane & 15;
    const int koff = (lane < 16) ? 0 : 2;
    const int hi   = (lane < 16) ? 0 : 8;
    const int q0   = blockIdx.x * 128 + wv * 16;

    const float* Qp = Q + (size_t)bh * (2048u * 64u);
    const float* Kp = K + (size_t)bh * (2048u * 64u);
    const float* Vp = V + (size_t)bh * (2048u * 64u);

    // Resident Q fragments (16 rows x 64 dims), pre-scaled by 1/sqrt(dk)
    v2f aq[16];
    #pragma unroll
    for (int kk = 0; kk < 16; ++kk) {
        const float* qr = &Qp[(size_t)(q0 + l16) * 64u + kk * 4 + koff];
        aq[kk].x = qr[0] * 0.125f;
        aq[kk].y = qr[1] * 0.125f;
    }

    // staging: 4 float4 per thread per matrix
    const int srow = tid >> 4;          // 0..15 (+16*i)
    const int scol = (tid & 15) * 4;    // 0..60

    float4 pk[4], pv[4];
    #pragma unroll
    for (int i = 0; i < 4; ++i) {
        pk[i] = *(const float4*)&Kp[(size_t)(srow + i * 16) * 64u + scol];
        pv[i] = *(const float4*)&Vp[(size_t)(srow + i * 16) * 64u + scol];
    }
    #pragma unroll
    for (int i = 0; i < 4; ++i) {
        int r = srow + i * 16;
        Ks[0][r][scol+0]=pk[i].x; Ks[0][r][scol+1]=pk[i].y;
        Ks[0][r][scol+2]=pk[i].z; Ks[0][r][scol+3]=pk[i].w;
        Vs[0][r][scol+0]=pv[i].x; Vs[0][r][scol+1]=pv[i].y;
        Vs[0][r][scol+2]=pv[i].z; Vs[0][r][scol+3]=pv[i].w;
    }
    __syncthreads();

    float m_run[8], l_run[8];
    v8f accO[4] = {};
    #pragma unroll
    for (int j = 0; j < 8; ++j) { m_run[j] = -1e30f; l_run[j] = 0.0f; }

    int buf = 0;
    for (int kb = 0; kb < 32; ++kb) {
        const bool notlast = (kb + 1 < 32);
        if (notlast) {   // prefetch next K/V block
            #pragma unroll
            for (int i = 0; i < 4; ++i) {
                pk[i] = *(const float4*)&Kp[(size_t)((kb+1)*64 + srow + i*16) * 64u + scol];
                pv[i] = *(const float4*)&Vp[(size_t)((kb+1)*64 + srow + i*16) * 64u + scol];
            }
        }

        // ---- S = (Q/sqrt(dk)) @ K^T : 4 chunks of 4 k-steps -------------
        v8f s[4] = {};
        #pragma unroll
        for (int kc = 0; kc < 16; kc += 4) {
            v2f bfr[4][4];
            #pragma unroll
            for (int kk = 0; kk < 4; ++kk)
                #pragma unroll
                for (int t = 0; t < 4; ++t) {
                    bfr[kk][t].x = Ks[buf][t*16 + l16][(kc+kk)*4 + koff];
                    bfr[kk][t].y = Ks[buf][t*16 + l16][(kc+kk)*4 + koff + 1];
                }
            #pragma unroll
            for (int kk = 0; kk < 4; ++kk)
                #pragma unroll
                for (int t = 0; t < 4; ++t)
                    s[t] = __builtin_amdgcn_wmma_f32_16x16x4_f32(
                        false, aq[kc+kk], false, bfr[kk][t], (short)0, s[t], false, false);
        }

        // ---- online softmax (row j in VGPR j; xor 1/2/4/8 stays in half) -
        #pragma unroll
        for (int j = 0; j < 8; ++j) {
            float mx = fmaxf(fmaxf(s[0][j], s[1][j]), fmaxf(s[2][j], s[3][j]));
            mx = fmaxf(mx, __shfl_xor(mx, 1, 32));
            mx = fmaxf(mx, __shfl_xor(mx, 2, 32));
            mx = fmaxf(mx, __shfl_xor(mx, 4, 32));
            mx = fmaxf(mx, __shfl_xor(mx, 8, 32));
            float mnew  = fmaxf(m_run[j], mx);
            float scale = __expf(m_run[j] - mnew);
            m_run[j] = mnew;
            float rs = 0.0f;
            #pragma unroll
            for (int t = 0; t < 4; ++t) {
                float p = __expf(s[t][j] - mnew);
                s[t][j] = p;
                rs += p;
            }
            rs += __shfl_xor(rs, 1, 32);
            rs += __shfl_xor(rs, 2, 32);
            rs += __shfl_xor(rs, 4, 32);
            rs += __shfl_xor(rs, 8, 32);
            l_run[j] = l_run[j] * scale + rs;
            #pragma unroll
            for (int t = 0; t < 4; ++t) accO[t][j] *= scale;
        }

        // ---- relayout P into per-wave LDS (same-wave, no barrier) -------
        #pragma unroll
        for (int t = 0; t < 4; ++t)
            #pragma unroll
            for (int j = 0; j < 8; ++j)
                Ps[wv][j + hi][t * 16 + l16] = s[t][j];

        // ---- O += P @ V : 4 chunks of 4 k-steps --------------------------
        #pragma unroll
        for (int kc = 0; kc < 16; kc += 4) {
            v2f afr[4], bfr[4][4];
            #pragma unroll
            for (int kk = 0; kk < 4; ++kk) {
                afr[kk].x = Ps[wv][l16][(kc+kk)*4 + koff];
                afr[kk].y = Ps[wv][l16][(kc+kk)*4 + koff + 1];
                #pragma unroll
                for (int t = 0; t < 4; ++t) {
                    bfr[kk][t].x = Vs[buf][(kc+kk)*4 + koff    ][t*16 + l16];
                    bfr[kk][t].y = Vs[buf][(kc+kk)*4 + koff + 1][t*16 + l16];
                }
            }
            #pragma unroll
            for (int kk = 0; kk < 4; ++kk)
                #pragma unroll
                for (int t = 0; t < 4; ++t)
                    accO[t] = __builtin_amdgcn_wmma_f32_16x16x4_f32(
                        false, afr[kk], false, bfr[kk][t], (short)0, accO[t], false, false);
        }

        if (notlast) {
            const int nb = buf ^ 1;
            #pragma unroll
            for (int i = 0; i < 4; ++i) {
                int r = srow + i * 16;
                Ks[nb][r][scol+0]=pk[i].x; Ks[nb][r][scol+1]=pk[i].y;
                Ks[nb][r][scol+2]=pk[i].z; Ks[nb][r][scol+3]=pk[i].w;
                Vs[nb][r][scol+0]=pv[i].x; Vs[nb][r][scol+1]=pv[i].y;
                Vs[nb][r][scol+2]=pv[i].z; Vs[nb][r][scol+3]=pv[i].w;
            }
            __syncthreads();
            buf = nb;
        }
    }

    // Normalize and store as (b, n, h*64+d)
    #pragma unroll
    for (int t = 0; t < 4; ++t) {
        #pragma unroll
        for (int j = 0; j < 8; ++j) {
            int row = q0 + j + hi;
            int d   = t * 16 + l16;
            O[((size_t)(b * 2048 + row)) * 1024u + h * 64 + d] = accO[t][j] / l_run[j];
        }
    }
}

// ---------------------------------------------------------------------------
// Workspace (floats): Weff x4 | Q | K | V | attn_out   = 80 MB total
// ---------------------------------------------------------------------------
extern "C" void kernel_launch(void* const* d_in, const int* in_sizes, int n_in,
                              void* d_out, int out_size, void* d_ws, size_t ws_size,
                              hipStream_t stream) {
    const float* x     = (const float*)d_in[0];
    const float* Wq_w  = (const float*)d_in[1];
    const float* Wq_dW = (const float*)d_in[2];
    const float* Wq_dB = (const float*)d_in[3];
    const float* Wk_w  = (const float*)d_in[4];
    const float* Wk_dW = (const float*)d_in[5];
    const float* Wk_dB = (const float*)d_in[6];
    const float* Wv_w  = (const float*)d_in[7];
    const float* Wv_dW = (const float*)d_in[8];
    const float* Wv_dB = (const float*)d_in[9];
    const float* Wo_w  = (const float*)d_in[10];
    const float* Wo_dW = (const float*)d_in[11];
    const float* Wo_dB = (const float*)d_in[12];

    float* ws   = (float*)d_ws;
    float* Weff = ws;
    float* qb   = ws + 4ull * 1048576ull;   // q,k,v contiguous (gemm z-offset relies on it)
    float* ao   = ws + 16ull * 1048576ull;
    float* out  = (float*)d_out;

    qalora_prep_weights<<<4096, 256, 0, stream>>>(
        Wq_w, Wq_dW, Wq_dB, Wk_w, Wk_dW, Wk_dB,
        Wv_w, Wv_dW, Wv_dB, Wo_w, Wo_dW, Wo_dB, Weff);

    // Fused Q/K/V projections: z = 0,1,2  (block tile 256x64 -> grid y = 16)
    gemm_weff_wmma<<<dim3(16, 16, 3), 256, 0, stream>>>(x, Weff, qb, /*out_mode=*/1);

    flash_attn_wmma<<<dim3(16, 32), 256, 0, stream>>>(
        qb, qb + 4ull * 1048576ull, qb + 8ull * 1048576ull, ao);

    // Output projection -> d_out
    gemm_weff_wmma<<<dim3(16, 16, 1), 256, 0, stream>>>(
        ao, Weff + 3ull * 1048576ull, out, /*out_mode=*/0);
}